// WeightScoreLayer_24283745091812
// MI455X (gfx1250) — compile-verified
//
#include <hip/hip_runtime.h>
#include <math.h>

// ---------------------------------------------------------------------------
// WeightScoreLayer for MI455X (gfx1250, wave32)
//   K0: zero workspace (deg + sum1 + sum2)                 [grid-stride f4]
//   K1: edge scatter: sum1[row] += x[col]; deg[row] += 1   [wave per edge]
//   K2: sum1 *= 1/max(deg,1)  (in-place -> x_mean)
//   K3: edge scatter: sum2[row] += |x[col] - mean[col]|    [wave per edge]
//   K4: score = sigmoid(feat . w) via V_WMMA_F32_16X16X4_F32, 16 nodes/wave
//
// Roofline: x (25.6 MB) is L2-resident (192 MB L2), so both edge passes are
// L2-bound scatter/gather (2 x E x D = 204.8M f32 atomics); HBM traffic is
// ~35 MB (~1.5us floor at 23.3 TB/s).  FLOPs are trivial, so f32 throughout;
// the final [N,384]x[384] GEMV runs on the matrix pipe.
// ---------------------------------------------------------------------------

typedef __attribute__((ext_vector_type(2))) float v2f;
typedef __attribute__((ext_vector_type(8))) float v8f;

#define DEMB 128   // embedding size (in_sizes[3]/3 == 128)

__device__ __forceinline__ void atomic_add_f32(float* p, float v) {
  __hip_atomic_fetch_add(p, v, __ATOMIC_RELAXED, __HIP_MEMORY_SCOPE_AGENT);
}

__global__ void zero_ws_kernel(float4* __restrict__ p, long nvec) {
  long i = (long)blockIdx.x * blockDim.x + threadIdx.x;
  long stride = (long)gridDim.x * blockDim.x;
  float4 z = {0.f, 0.f, 0.f, 0.f};
  for (long j = i; j < nvec; j += stride) p[j] = z;
}

// Pass 1: one wave32 per edge; lane l moves x[col][4l..4l+3] (coalesced 512B)
__global__ void agg_pass1_kernel(const float* __restrict__ x,
                                 const int* __restrict__ row,
                                 const int* __restrict__ col,
                                 float* __restrict__ sum1,
                                 float* __restrict__ deg, int E) {
  int edge = blockIdx.x * (blockDim.x >> 5) + (threadIdx.x >> 5);
  if (edge >= E) return;
  int lane = threadIdx.x & 31;
  int r = row[edge];
  int c = col[edge];
  float4 v = ((const float4*)(x + (long)c * DEMB))[lane];
  float* dst = sum1 + (long)r * DEMB + lane * 4;
  atomic_add_f32(dst + 0, v.x);
  atomic_add_f32(dst + 1, v.y);
  atomic_add_f32(dst + 2, v.z);
  atomic_add_f32(dst + 3, v.w);
  if (lane == 0) atomic_add_f32(deg + r, 1.0f);
}

// In-place normalize: sum1 -> x_mean.  One thread per float4 chunk (32/node).
__global__ void normalize_kernel(float* __restrict__ sum1,
                                 const float* __restrict__ deg, int n) {
  long idx = (long)blockIdx.x * blockDim.x + threadIdx.x;
  long total = (long)n * (DEMB / 4);
  if (idx >= total) return;
  int node = (int)(idx >> 5);                 // DEMB/4 == 32 chunks per node
  float inv = 1.0f / fmaxf(deg[node], 1.0f);
  float4* p = (float4*)sum1 + idx;
  float4 v = *p;
  v.x *= inv; v.y *= inv; v.z *= inv; v.w *= inv;
  *p = v;
}

// Pass 2: recompute |x - mean| on the fly (both L2-resident), scatter-add.
__global__ void agg_pass2_kernel(const float* __restrict__ x,
                                 const float* __restrict__ mean,
                                 const int* __restrict__ row,
                                 const int* __restrict__ col,
                                 float* __restrict__ sum2, int E) {
  int edge = blockIdx.x * (blockDim.x >> 5) + (threadIdx.x >> 5);
  if (edge >= E) return;
  int lane = threadIdx.x & 31;
  int r = row[edge];
  int c = col[edge];
  long off = (long)c * DEMB;
  float4 xv = ((const float4*)(x + off))[lane];
  float4 mv = ((const float4*)(mean + off))[lane];
  float* dst = sum2 + (long)r * DEMB + lane * 4;
  atomic_add_f32(dst + 0, fabsf(xv.x - mv.x));
  atomic_add_f32(dst + 1, fabsf(xv.y - mv.y));
  atomic_add_f32(dst + 2, fabsf(xv.z - mv.z));
  atomic_add_f32(dst + 3, fabsf(xv.w - mv.w));
}

// K4: one wave32 computes 16 node scores with V_WMMA_F32_16X16X4_F32.
// A (16x4 f32, ISA layout): lanes 0-15 = rows M=0..15, VGPR0 = K{0|2 by half},
// VGPR1 = K{1|3}.  B (4x16): row k is constant (=w[k]) across all 16 columns,
// so each lane just needs its K-pair of w — same indexing as A.
// D row M depends only on A row M, so out-of-range rows need no masking
// (their stores are guarded); only the addresses are clamped.
// Loops run over a uniform counter i (lane-dependent kbase folded into the
// base pointers) so the trip count is a compile-time constant -> full unroll,
// EXEC stays all-1s, and loads use immediate offsets off one base address.
// After 96 accumulating WMMAs, D VGPR r holds row r (lanes 0-15) / 8+r
// (lanes 16-31), identical across N-columns; lanes 0 and 16 write out.
__global__ void score_wmma_kernel(const float* __restrict__ x,
                                  const float* __restrict__ mean,
                                  const float* __restrict__ sum2,
                                  const float* __restrict__ deg,
                                  const float* __restrict__ w,
                                  float* __restrict__ out, int n) {
  int lane = threadIdx.x & 31;
  int waveGlobal = blockIdx.x * (blockDim.x >> 5) + (threadIdx.x >> 5);
  int base = waveGlobal * 16;
  int half = lane >> 4;       // 0: lanes 0-15, 1: lanes 16-31
  int mrel = lane & 15;
  int m = base + mrel;
  int mc = m < n ? m : (n - 1);              // clamp for address safety only
  long rowOff = (long)mc * DEMB;
  float inv = 1.0f / fmaxf(deg[mc], 1.0f);

  int kbase = 2 * half;       // this lane's K-pair start within each chunk of 4
  const float* pxk = x    + rowOff + kbase;
  const float* pmk = mean + rowOff + kbase;
  const float* psk = sum2 + rowOff + kbase;
  const float* w0k = w + kbase;
  const float* w1k = w + DEMB + kbase;
  const float* w2k = w + 2 * DEMB + kbase;

  v8f c = {};
  // region 0: feat = x_mean * x
#pragma unroll
  for (int i = 0; i < DEMB / 4; ++i) {
    v2f a = (*(const v2f*)(pmk + 4 * i)) * (*(const v2f*)(pxk + 4 * i));
    v2f b = *(const v2f*)(w0k + 4 * i);
    c = __builtin_amdgcn_wmma_f32_16x16x4_f32(false, a, false, b,
                                              (short)0, c, false, false);
  }
  // region 1: feat = x_std = sum2 * inv_deg
#pragma unroll
  for (int i = 0; i < DEMB / 4; ++i) {
    v2f a = (*(const v2f*)(psk + 4 * i)) * inv;
    v2f b = *(const v2f*)(w1k + 4 * i);
    c = __builtin_amdgcn_wmma_f32_16x16x4_f32(false, a, false, b,
                                              (short)0, c, false, false);
  }
  // region 2: feat = x
#pragma unroll
  for (int i = 0; i < DEMB / 4; ++i) {
    v2f a = *(const v2f*)(pxk + 4 * i);
    v2f b = *(const v2f*)(w2k + 4 * i);
    c = __builtin_amdgcn_wmma_f32_16x16x4_f32(false, a, false, b,
                                              (short)0, c, false, false);
  }

  if (mrel == 0) {            // lane 0 -> rows 0..7, lane 16 -> rows 8..15
    for (int r = 0; r < 8; ++r) {
      int node = base + half * 8 + r;
      if (node < n) {
        float v = c[r];
        out[node] = 1.0f / (1.0f + expf(-v));
      }
    }
  }
}

extern "C" void kernel_launch(void* const* d_in, const int* in_sizes, int n_in,
                              void* d_out, int out_size, void* d_ws, size_t ws_size,
                              hipStream_t stream) {
  const float* x   = (const float*)d_in[0];   // [N, 128]
  const int*   row = (const int*)d_in[1];     // [E]
  const int*   col = (const int*)d_in[2];     // [E]
  const float* w   = (const float*)d_in[3];   // [384]
  float* out = (float*)d_out;                 // [N]

  const int N = out_size;                     // 50000
  const int E = in_sizes[1];                  // 800000

  // Workspace layout (floats): deg[N] | sum1/mean[N*D] | sum2[N*D]  (~51.4 MB)
  float* deg  = (float*)d_ws;
  float* sum1 = deg + N;                      // N*4 bytes is 16B-aligned (N%4==0)
  float* sum2 = sum1 + (long)N * DEMB;

  // K0: zero accumulators (must run every call — atomics accumulate).
  long zeroVec = ((long)N + 2L * N * DEMB) / 4;
  zero_ws_kernel<<<4096, 256, 0, stream>>>((float4*)deg, zeroVec);

  // K1 / K3: one wave32 per edge, 8 waves per 256-thread block.
  int edgeBlocks = (E + 7) / 8;
  agg_pass1_kernel<<<edgeBlocks, 256, 0, stream>>>(x, row, col, sum1, deg, E);

  long chunks = (long)N * (DEMB / 4);
  int normBlocks = (int)((chunks + 255) / 256);
  normalize_kernel<<<normBlocks, 256, 0, stream>>>(sum1, deg, N);

  agg_pass2_kernel<<<edgeBlocks, 256, 0, stream>>>(x, sum1, row, col, sum2, E);

  // K4: 16 nodes per wave, 8 waves per block.
  int waves = (N + 15) / 16;
  int scoreBlocks = (waves + 7) / 8;
  score_wmma_kernel<<<scoreBlocks, 256, 0, stream>>>(x, sum1, sum2, deg, w, out, N);
}